// GNN_32590211842293
// MI455X (gfx1250) — compile-verified
//
#include <hip/hip_runtime.h>
#include <hip/hip_bf16.h>
#include <math.h>

#define N_NODES  100000
#define N_EDGES  1200000
#define HID      64
#define N_GRAPHS 128
#define N_CLASSES 3

typedef __attribute__((ext_vector_type(2))) float v2f;
typedef __attribute__((ext_vector_type(8))) float v8f;

// ---------------- zero fill ----------------
__global__ void gnn_zero(float* __restrict__ p, long long n) {
    long long i = (long long)blockIdx.x * blockDim.x + threadIdx.x;
    if (i < n) p[i] = 0.0f;
}

// ---------------- edge scatter-add: agg[dst] += x[src] ----------------
// thread = (edge, feature). 64 consecutive threads share an edge, so the
// gather of row x[src] (256B) and the atomic adds to row agg[dst] are coalesced.
__global__ __launch_bounds__(256) void gnn_edge_scatter(
    const float* __restrict__ xin,
    const int* __restrict__ src, const int* __restrict__ dst,
    float* __restrict__ agg, int nEdges)
{
    long long gid = (long long)blockIdx.x * blockDim.x + threadIdx.x;
    int e = (int)(gid >> 6);
    int f = (int)(gid & 63);
    if (e < nEdges) {
        int s = src[e];
        int d = dst[e];
        float v = xin[(long long)s * HID + f];
        unsafeAtomicAdd(&agg[(long long)d * HID + f], v);  // global_atomic_add_f32
    }
}

// ---------------- GraphConv GEMM via fp32 WMMA ----------------
// out[m,n] = relu( sum_k agg[m,k]*Wr[k,n] + sum_k x[m,k]*Ws[k,n] + br[n] )
// One wave computes one 16x16 output tile, K=64 via 16 steps of
// v_wmma_f32_16x16x4_f32 per input matrix (exact fp32 — workload is
// bandwidth-bound, so no reason to downcast).
//
// A-frag (16x4 f32): lane L<16 holds row M=L, K = 4*ks + {0,1};
//                    lane L>=16 holds row M=L-16, K = 4*ks + {2,3}.
// B-frag (4x16 f32): lane L<16 holds col N=L, K = 4*ks + {0,1} in v0/v1;
//                    lane L>=16 holds col N=L-16, K = 4*ks + {2,3}.
// C/D (16x16 f32, 8 VGPRs): elem r -> M = r + 8*(lane>=16), N = lane&15.
__global__ __launch_bounds__(128) void gnn_conv_wmma(
    const float* __restrict__ agg, const float* __restrict__ xin,
    const float* __restrict__ Wr,  const float* __restrict__ br,
    const float* __restrict__ Ws,  float* __restrict__ out, int nRows)
{
    const int lane    = threadIdx.x & 31;
    const int wave    = threadIdx.x >> 5;
    const int tile    = blockIdx.x * 4 + wave;
    const int rowTile = tile >> 2;          // HID=64 -> 4 column tiles
    const int colTile = tile & 3;
    if (rowTile * 16 >= nRows) return;      // wave-uniform guard (EXEC stays all-1)

    const int m     = rowTile * 16 + (lane & 15);
    const int n     = colTile * 16 + (lane & 15);
    const int khalf = (lane >> 4) * 2;      // 0 or 2

    const float* aRow = agg + (long long)m * HID;
    const float* xRow = xin + (long long)m * HID;

    v8f c = {0.f, 0.f, 0.f, 0.f, 0.f, 0.f, 0.f, 0.f};

#pragma unroll
    for (int ks = 0; ks < 16; ++ks) {
        const int k0 = ks * 4 + khalf;
        v2f a = *(const v2f*)(aRow + k0);                 // 8B aligned
        v2f b;
        b.x = Wr[(k0    ) * HID + n];
        b.y = Wr[(k0 + 1) * HID + n];
        c = __builtin_amdgcn_wmma_f32_16x16x4_f32(false, a, false, b,
                                                  (short)0, c, false, false);
    }
#pragma unroll
    for (int ks = 0; ks < 16; ++ks) {
        const int k0 = ks * 4 + khalf;
        v2f a = *(const v2f*)(xRow + k0);
        v2f b;
        b.x = Ws[(k0    ) * HID + n];
        b.y = Ws[(k0 + 1) * HID + n];
        c = __builtin_amdgcn_wmma_f32_16x16x4_f32(false, a, false, b,
                                                  (short)0, c, false, false);
    }

    const float bias  = br[n];
    const int   mBase = rowTile * 16 + (lane >> 4) * 8;
#pragma unroll
    for (int r = 0; r < 8; ++r) {
        float v = c[r] + bias;
        v = fmaxf(v, 0.0f);                 // all three conv layers apply ReLU
        out[(long long)(mBase + r) * HID + n] = v;
    }
}

// ---------------- global_add_pool over sorted batch ids ----------------
__global__ __launch_bounds__(64) void gnn_pool(
    const float* __restrict__ h, const int* __restrict__ batch,
    float* __restrict__ g, int nNodes, int nodesPerBlock)
{
    const int f     = threadIdx.x;                      // 0..63
    const int start = blockIdx.x * nodesPerBlock;
    int end = start + nodesPerBlock;
    if (end > nNodes) end = nNodes;
    if (start >= nNodes) return;

    float acc = 0.0f;
    int   cur = -1;
    for (int nd = start; nd < end; ++nd) {
        int b = batch[nd];                              // sorted -> long runs
        if (b != cur) {
            if (cur >= 0) unsafeAtomicAdd(&g[(long long)cur * HID + f], acc);
            acc = 0.0f;
            cur = b;
        }
        acc += h[(long long)nd * HID + f];
    }
    if (cur >= 0) unsafeAtomicAdd(&g[(long long)cur * HID + f], acc);
}

// ---------------- head: relu(g@W1+b1) @ W2 + b2, log_softmax ----------------
__global__ __launch_bounds__(64) void gnn_head(
    const float* __restrict__ g,
    const float* __restrict__ W1, const float* __restrict__ b1,
    const float* __restrict__ W2, const float* __restrict__ b2,
    float* __restrict__ out)
{
    __shared__ float sg[HID];
    __shared__ float sh[HID];
    __shared__ float logits[N_CLASSES];

    const int t  = threadIdx.x;
    const int gi = blockIdx.x;

    sg[t] = g[(long long)gi * HID + t];
    __syncthreads();

    float acc = b1[t];
#pragma unroll 8
    for (int k = 0; k < HID; ++k) acc += sg[k] * W1[k * HID + t];
    sh[t] = fmaxf(acc, 0.0f);
    __syncthreads();

    if (t < N_CLASSES) {
        float l = b2[t];
        for (int k = 0; k < HID; ++k) l += sh[k] * W2[k * N_CLASSES + t];
        logits[t] = l;
    }
    __syncthreads();

    if (t == 0) {
        float m = fmaxf(logits[0], fmaxf(logits[1], logits[2]));
        float s = expf(logits[0] - m) + expf(logits[1] - m) + expf(logits[2] - m);
        float lse = m + logf(s);
        out[(long long)gi * N_CLASSES + 0] = logits[0] - lse;
        out[(long long)gi * N_CLASSES + 1] = logits[1] - lse;
        out[(long long)gi * N_CLASSES + 2] = logits[2] - lse;
    }
}

// ---------------- host-side orchestration ----------------
extern "C" void kernel_launch(void* const* d_in, const int* in_sizes, int n_in,
                              void* d_out, int out_size, void* d_ws, size_t ws_size,
                              hipStream_t stream) {
    const float* x     = (const float*)d_in[0];
    const int*   ei    = (const int*)  d_in[1];
    const int*   batch = (const int*)  d_in[2];
    const float* Wr1 = (const float*)d_in[3];
    const float* br1 = (const float*)d_in[4];
    const float* Ws1 = (const float*)d_in[5];
    const float* Wr2 = (const float*)d_in[6];
    const float* br2 = (const float*)d_in[7];
    const float* Ws2 = (const float*)d_in[8];
    const float* Wr3 = (const float*)d_in[9];
    const float* br3 = (const float*)d_in[10];
    const float* Ws3 = (const float*)d_in[11];
    const float* Wf1 = (const float*)d_in[12];
    const float* bf1 = (const float*)d_in[13];
    const float* Wf2 = (const float*)d_in[14];
    const float* bf2 = (const float*)d_in[15];
    float* out = (float*)d_out;

    const int* srcIdx = ei;
    const int* dstIdx = ei + N_EDGES;

    const long long nodeElems = (long long)N_NODES * HID;     // 6.4M floats
    float* agg = (float*)d_ws;
    float* hA  = agg + nodeElems;
    float* hB  = hA + nodeElems;
    float* gP  = hB + nodeElems;                              // [128,64]

    const int zeroBlocks    = (int)((nodeElems + 255) / 256);
    const int scatterBlocks = (int)(((long long)N_EDGES * HID + 255) / 256);
    const int convBlocks    = (N_NODES + 63) / 64;            // 1563 blocks, 1 row-tile each (4 col waves)
    const int nodesPerBlk   = 256;
    const int poolBlocks    = (N_NODES + nodesPerBlk - 1) / nodesPerBlk;

    // ---- layer 1: agg = scatter(x); hA = relu(agg@Wr1 + x@Ws1 + br1)
    gnn_zero<<<zeroBlocks, 256, 0, stream>>>(agg, nodeElems);
    gnn_edge_scatter<<<scatterBlocks, 256, 0, stream>>>(x, srcIdx, dstIdx, agg, N_EDGES);
    gnn_conv_wmma<<<convBlocks, 128, 0, stream>>>(agg, x, Wr1, br1, Ws1, hA, N_NODES);

    // ---- layer 2
    gnn_zero<<<zeroBlocks, 256, 0, stream>>>(agg, nodeElems);
    gnn_edge_scatter<<<scatterBlocks, 256, 0, stream>>>(hA, srcIdx, dstIdx, agg, N_EDGES);
    gnn_conv_wmma<<<convBlocks, 128, 0, stream>>>(agg, hA, Wr2, br2, Ws2, hB, N_NODES);

    // ---- layer 3
    gnn_zero<<<zeroBlocks, 256, 0, stream>>>(agg, nodeElems);
    gnn_edge_scatter<<<scatterBlocks, 256, 0, stream>>>(hB, srcIdx, dstIdx, agg, N_EDGES);
    gnn_conv_wmma<<<convBlocks, 128, 0, stream>>>(agg, hB, Wr3, br3, Ws3, hA, N_NODES);

    // ---- pool + head
    gnn_zero<<<(N_GRAPHS * HID + 255) / 256, 256, 0, stream>>>(gP, (long long)N_GRAPHS * HID);
    gnn_pool<<<poolBlocks, 64, 0, stream>>>(hA, batch, gP, N_NODES, nodesPerBlk);
    gnn_head<<<N_GRAPHS, 64, 0, stream>>>(gP, Wf1, bf1, Wf2, bf2, out);
}